// Sampler_52467320488051
// MI455X (gfx1250) — compile-verified
//
#include <hip/hip_runtime.h>
#include <stdint.h>

#define N_ROWS 256
#define VOCAB  50257
#define P_LEN  512
#define O_LEN  128
#define BLK    1024
#define NWAVE  32
#define OBIN_W 12565   /* ceil(VOCAB/4)  bytes packed in u32 */
#define SEEN_W 1571    /* ceil(VOCAB/32) bit set             */
#define SMEM_W 14136   /* == OBIN_W+SEEN_W, also >= 8448+256 for radix counters */

#define NEG_INF (-__builtin_huge_valf())

// ---------------- float <-> sortable key (monotone bijection) ----------------
__device__ __forceinline__ uint32_t f2k(float f) {
  uint32_t u = __float_as_uint(f);
  return u ^ (uint32_t)(((int32_t)u >> 31) | 0x80000000);
}
__device__ __forceinline__ float k2f(uint32_t k) {
  uint32_t u = (k & 0x80000000u) ? (k & 0x7FFFFFFFu) : ~k;
  return __uint_as_float(u);
}

// ---------------- CDNA5 async global->LDS (ASYNCcnt path) --------------------
#if defined(__has_builtin)
#if __has_builtin(__builtin_amdgcn_global_load_async_to_lds_b32)
#define HAS_ASYNC_BUILTIN 1
#endif
#if __has_builtin(__builtin_amdgcn_s_wait_asynccnt)
#define HAS_WAITASYNC_BUILTIN 1
#endif
#endif

__device__ __forceinline__ void async_g2l_b32(uint32_t* lds_dst, const uint32_t* gsrc) {
#ifdef HAS_ASYNC_BUILTIN
  typedef __attribute__((address_space(1))) int gint_t;   // global
  typedef __attribute__((address_space(3))) int lint_t;   // LDS
  __builtin_amdgcn_global_load_async_to_lds_b32(
      (gint_t*)(uintptr_t)gsrc,
      (lint_t*)(uint32_t)(uintptr_t)lds_dst, 0, 0);
#else
  uint32_t loff = (uint32_t)(uintptr_t)lds_dst;  // low 32 bits of flat LDS ptr = LDS offset
  asm volatile("global_load_async_to_lds_b32 %0, %1, off"
               :: "v"(loff), "v"(gsrc) : "memory");
#endif
}
__device__ __forceinline__ void wait_asynccnt0() {
#ifdef HAS_WAITASYNC_BUILTIN
  __builtin_amdgcn_s_wait_asynccnt(0);
#else
  asm volatile("s_wait_asynccnt 0" ::: "memory");
#endif
}
__device__ __forceinline__ void wait_asynccnt1() {   // allow 1 outstanding (double buffer)
#ifdef HAS_WAITASYNC_BUILTIN
  __builtin_amdgcn_s_wait_asynccnt(1);
#else
  asm volatile("s_wait_asynccnt 1" ::: "memory");
#endif
}

// ---------------- block-wide primitives (wave32) -----------------------------
__device__ __forceinline__ float block_reduce_add(float x, float* wsum, int tid) {
  int lane = tid & 31, wid = tid >> 5;
  for (int off = 16; off > 0; off >>= 1) x += __shfl_down(x, off, 32);
  if (lane == 0) wsum[wid] = x;
  __syncthreads();
  if (tid == 0) { float r = 0.f; for (int w = 0; w < NWAVE; ++w) r += wsum[w]; wsum[NWAVE] = r; }
  __syncthreads();
  float r = wsum[NWAVE];
  __syncthreads();
  return r;
}

__device__ __forceinline__ float block_scan_inclusive(float x, float* wsum, int tid, float* total) {
  int lane = tid & 31, wid = tid >> 5;
  for (int off = 1; off < 32; off <<= 1) {
    float y = __shfl_up(x, off, 32);
    if (lane >= off) x += y;
  }
  if (lane == 31) wsum[wid] = x;
  __syncthreads();
  if (wid == 0) {
    float w = wsum[lane];
    for (int off = 1; off < 32; off <<= 1) {
      float y = __shfl_up(w, off, 32);
      if (lane >= off) w += y;
    }
    wsum[lane] = w;
  }
  __syncthreads();
  float base = wid ? wsum[wid - 1] : 0.0f;
  *total = wsum[NWAVE - 1];
  return x + base;
}

// binary searches over sorted ascending keys
__device__ __forceinline__ int lb_idx(const uint32_t* S, uint32_t k) {  // first i: S[i] >= k
  int lo = 0, hi = VOCAB;
  while (lo < hi) { int mid = (lo + hi) >> 1; if (S[mid] < k) lo = mid + 1; else hi = mid; }
  return lo;
}
__device__ __forceinline__ int ub_idx(const uint32_t* S, uint32_t k) {  // first i: S[i] > k
  int lo = 0, hi = VOCAB;
  while (lo < hi) { int mid = (lo + hi) >> 1; if (S[mid] <= k) lo = mid + 1; else hi = mid; }
  return lo;
}
// mass of survivors with value in [A-d, A+d]  (P = inclusive prefix of probs)
__device__ __forceinline__ float w_interval(const uint32_t* S, const float* P, float A, float d) {
  int hi = ub_idx(S, f2k(A + d));
  int lo = lb_idx(S, f2k(A - d));
  float shi = (hi > 0) ? P[hi - 1] : 0.0f;
  float slo = (lo > 0) ? P[lo - 1] : 0.0f;
  return shi - slo;
}

// =============================================================================
__global__ __launch_bounds__(BLK)
void sampler_row_kernel(const float* __restrict__ logits,
                        const int*   __restrict__ prompt_tok,
                        const int*   __restrict__ output_tok,
                        const float* __restrict__ presence_p,
                        const float* __restrict__ freq_p,
                        const float* __restrict__ rep_p,
                        const float* __restrict__ temps,
                        const float* __restrict__ top_p,
                        const int*   __restrict__ top_k,
                        const float* __restrict__ min_p,
                        const float* __restrict__ top_a,
                        const float* __restrict__ tfs_p,
                        const float* __restrict__ typ_p,
                        const float* __restrict__ eps_cut,
                        const float* __restrict__ eta_cut,
                        const float* __restrict__ smooth_f,
                        const float* __restrict__ smooth_c,
                        float* __restrict__ out,
                        uint32_t* __restrict__ wsA,
                        uint32_t* __restrict__ wsB) {
  const int row  = blockIdx.x;
  const int tid  = threadIdx.x;
  const int lane = tid & 31;
  const int wid  = tid >> 5;

  const float* lrow = logits + (size_t)row * VOCAB;
  float*       orow = out    + (size_t)row * VOCAB;
  uint32_t*    A    = wsA    + (size_t)row * VOCAB;
  uint32_t*    B    = wsB    + (size_t)row * VOCAB;

  __shared__ uint32_t s_mem[SMEM_W];
  __shared__ uint32_t s_stage[2 * BLK];     // double-buffered async staging
  __shared__ float    s_wsum[NWAVE + 1];
  __shared__ float    s_f[4];
  __shared__ int      s_i[4];
  __shared__ uint32_t s_u[2];

  const float pres = presence_p[row];
  const float freq = freq_p[row];
  const float rep  = rep_p[row];
  const float invT = 1.0f / temps[row];
  const float topp = top_p[row];
  const int   topk = top_k[row];
  const float minp = min_p[row];
  const float topa = top_a[row];
  const float tfsr = tfs_p[row];
  const float typr = typ_p[row];
  const float epsr = eps_cut[row];
  const float etar = eta_cut[row];
  const float smf  = smooth_f[row];
  const float smcv = smooth_c[row];

  // ---------------- Phase 1: token bins in LDS -------------------------------
  uint32_t* obin = s_mem;            // VOCAB u8 counts, packed
  uint32_t* seen = s_mem + OBIN_W;   // VOCAB bit flags
  for (int i = tid; i < OBIN_W + SEEN_W; i += BLK) s_mem[i] = 0u;
  __syncthreads();
  for (int i = tid; i < P_LEN; i += BLK) {
    int t = prompt_tok[row * P_LEN + i];
    if ((unsigned)t < VOCAB) atomicOr(&seen[t >> 5], 1u << (t & 31));
  }
  for (int i = tid; i < O_LEN; i += BLK) {
    int t = output_tok[row * O_LEN + i];
    if ((unsigned)t < VOCAB) {
      atomicOr(&seen[t >> 5], 1u << (t & 31));
      atomicAdd(&obin[t >> 2], 1u << ((t & 3) * 8));
    }
  }
  __syncthreads();

  // ---------------- Phase 2: penalties + temperature; emit keys --------------
  for (int i = tid; i < VOCAB; i += BLK) {
    __builtin_prefetch(lrow + i + 4 * BLK, 0, 0);   // global_prefetch_b8
    float l = lrow[i];
    uint32_t c = (obin[i >> 2] >> ((i & 3) * 8)) & 0xFFu;
    bool sn = (seen[i >> 5] >> (i & 31)) & 1u;
    float r = sn ? rep : 1.0f;
    l = (l > 0.0f) ? (l / r) : (l * r);
    l -= freq * (float)c;
    l -= pres * (c ? 1.0f : 0.0f);
    l *= invT;
    orow[i] = l;
    A[i]    = f2k(l);
  }
  __threadfence();
  __syncthreads();

  // ---------------- Phase 3: stable LSD radix sort (4x8-bit), keys only ------
  {
    uint32_t* wavecnt = s_mem;          // [32][256]
    uint32_t* runoff  = s_mem + 8192;   // 256 (also the histogram)
    uint32_t* ctot    = s_mem + 8448;   // 256
    const int nchunk  = (VOCAB + BLK - 1) / BLK;
    for (int pass = 0; pass < 4; ++pass) {
      const uint32_t* src = (pass & 1) ? B : A;
      uint32_t*       dst = (pass & 1) ? A : B;
      const int shift = pass * 8;

      // global digit histogram: async global->LDS staging, double-buffered.
      // Each thread's staging slot is private to its wave, so the only sync
      // needed is the per-wave ASYNCcnt wait: issue tile n+1, wait <=1, eat tile n.
      for (int i = tid; i < 256; i += BLK) runoff[i] = 0u;
      __syncthreads();
      async_g2l_b32(&s_stage[tid], &src[tid]);           // tile 0 (VOCAB > BLK)
      for (int ch = 0; ch < nchunk; ++ch) {
        if (ch + 1 < nchunk) {
          int inext = (ch + 1) * BLK + tid;
          if (inext < VOCAB)
            async_g2l_b32(&s_stage[((ch + 1) & 1) * BLK + tid], &src[inext]);
          wait_asynccnt1();
        } else {
          wait_asynccnt0();
        }
        int i = ch * BLK + tid;
        if (i < VOCAB)
          atomicAdd(&runoff[(s_stage[(ch & 1) * BLK + tid] >> shift) & 255u], 1u);
      }
      __syncthreads();
      if (tid == 0) {                 // exclusive scan of 256 bins
        uint32_t sum = 0;
        for (int b = 0; b < 256; ++b) { uint32_t c = runoff[b]; runoff[b] = sum; sum += c; }
      }
      __syncthreads();

      // stable scatter, chunk by chunk
      for (int base = 0; base < VOCAB; base += BLK) {
        for (int j = tid; j < NWAVE * 256; j += BLK) wavecnt[j] = 0u;
        __syncthreads();
        int i = base + tid;
        bool valid = i < VOCAB;
        uint32_t key = valid ? src[i] : 0u;
        uint32_t dgt = (key >> shift) & 255u;
        uint32_t active = (uint32_t)__ballot(valid);
        uint32_t peers  = active;
        for (int b = 0; b < 8; ++b) {
          uint32_t bal = (uint32_t)__ballot(valid && ((dgt >> b) & 1u));
          peers &= ((dgt >> b) & 1u) ? bal : ~bal;
        }
        int rank = __popc(peers & ((1u << lane) - 1u));
        if (valid && rank == 0) wavecnt[wid * 256 + dgt] = (uint32_t)__popc(peers);
        __syncthreads();
        if (tid < 256) {             // exclusive scan over waves, per digit
          uint32_t sum = 0;
          for (int w = 0; w < NWAVE; ++w) {
            uint32_t t = wavecnt[w * 256 + tid];
            wavecnt[w * 256 + tid] = sum;
            sum += t;
          }
          ctot[tid] = sum;
        }
        __syncthreads();
        if (valid) {
          uint32_t pos = runoff[dgt] + wavecnt[wid * 256 + dgt] + (uint32_t)rank;
          dst[pos] = key;
        }
        __syncthreads();
        if (tid < 256) runoff[tid] += ctot[tid];
        __syncthreads();
      }
      __threadfence();
      __syncthreads();
    }
  }
  const uint32_t* S = A;          // sorted ascending keys
  float* Pbuf = (float*)B;        // reused for prefix probs

  // ---------------- Phase 4: order statistics --------------------------------
  const uint32_t kmax_key = S[VOCAB - 1];
  const float m = k2f(kmax_key);
  int kk = topk; if (kk < 1) kk = 1; if (kk > VOCAB) kk = VOCAB;
  const uint32_t kkey = S[VOCAB - kk];

  // Z1 over top-k survivors
  float acc = 0.f;
  for (int i = tid; i < VOCAB; i += BLK) {
    uint32_t key = S[i];
    if (key >= kkey) acc += __expf(k2f(key) - m);
  }
  const float Z1 = block_reduce_add(acc, s_wsum, tid);

  // top-p: ascending weighted scan, cutoff index c
  if (tid == 0) s_i[0] = -1;
  __syncthreads();
  {
    const float target = (1.0f - topp) * Z1;
    float carry = 0.f;
    for (int base = 0; base < VOCAB; base += BLK) {
      int i = base + tid;
      float e = 0.f;
      if (i < VOCAB) { uint32_t key = S[i]; if (key >= kkey) e = __expf(k2f(key) - m); }
      float tot;
      float cs = carry + block_scan_inclusive(e, s_wsum, tid, &tot);
      if (i < VOCAB && cs <= target) atomicMax(&s_i[0], i);
      carry += tot;
      __syncthreads();
    }
  }
  int c = s_i[0]; if (c > VOCAB - 2) c = VOCAB - 2;
  const uint32_t tpkey  = (c >= 0) ? S[c + 1] : 0u;
  const uint32_t lowKey2 = (kkey > tpkey) ? kkey : tpkey;

  // TFS: second differences of descending (unnormalized) probs
  acc = 0.f;
  for (int j = tid; j < VOCAB - 2; j += BLK) {
    uint32_t k0 = S[VOCAB - 1 - j], k1 = S[VOCAB - 2 - j], k2 = S[VOCAB - 3 - j];
    float p0 = (k0 >= lowKey2) ? __expf(k2f(k0) - m) : 0.f;
    float p1 = (k1 >= lowKey2) ? __expf(k2f(k1) - m) : 0.f;
    float p2 = (k2 >= lowKey2) ? __expf(k2f(k2) - m) : 0.f;
    acc += fabsf(p2 - 2.f * p1 + p0);
  }
  const float sumd2 = block_reduce_add(acc, s_wsum, tid);
  if (tid == 0) s_i[1] = VOCAB - 2;
  __syncthreads();
  {
    const float thr = tfsr * sumd2;
    float carry = 0.f;
    for (int base = 0; base < VOCAB - 2; base += BLK) {
      int j = base + tid;
      float d2 = 0.f;
      if (j < VOCAB - 2) {
        uint32_t k0 = S[VOCAB - 1 - j], k1 = S[VOCAB - 2 - j], k2 = S[VOCAB - 3 - j];
        float p0 = (k0 >= lowKey2) ? __expf(k2f(k0) - m) : 0.f;
        float p1 = (k1 >= lowKey2) ? __expf(k2f(k1) - m) : 0.f;
        float p2 = (k2 >= lowKey2) ? __expf(k2f(k2) - m) : 0.f;
        d2 = fabsf(p2 - 2.f * p1 + p0);
      }
      float tot;
      float cum = carry + block_scan_inclusive(d2, s_wsum, tid, &tot);
      if (j < VOCAB - 2 && cum > thr) atomicMin(&s_i[1], j);
      carry += tot;
      __syncthreads();
    }
  }
  const int i0 = s_i[1];
  const uint32_t tfskey  = S[VOCAB - 1 - i0];
  const uint32_t lowKey3 = (lowKey2 > tfskey) ? lowKey2 : tfskey;

  // eta cutoff: Z3 + entropy
  float a0 = 0.f, a1 = 0.f;
  for (int i = tid; i < VOCAB; i += BLK) {
    uint32_t key = S[i];
    if (key >= lowKey3) { float v = k2f(key); float e = __expf(v - m); a0 += e; a1 += e * (v - m); }
  }
  const float Z3   = block_reduce_add(a0, s_wsum, tid);
  const float Sve3 = block_reduce_add(a1, s_wsum, tid);
  const float negent = Sve3 / Z3 - logf(Z3);
  const float eps_eta = fminf(etar, sqrtf(etar) * __expf(negent));
  uint32_t etakey = f2k(m + logf(eps_eta * Z3));
  if (etakey > kmax_key) etakey = kmax_key;            // always keep argmax
  const uint32_t lowKey4 = (lowKey3 > etakey) ? lowKey3 : etakey;

  // epsilon cutoff: Z4
  acc = 0.f;
  for (int i = tid; i < VOCAB; i += BLK) {
    uint32_t key = S[i];
    if (key >= lowKey4) acc += __expf(k2f(key) - m);
  }
  const float Z4 = block_reduce_add(acc, s_wsum, tid);
  uint32_t epskey = f2k(m + logf(epsr * Z4));
  if (epskey > kmax_key) epskey = kmax_key;
  const uint32_t lowKey5 = (lowKey4 > epskey) ? lowKey4 : epskey;

  // typical: Z5, entropy center A_val, prefix probs, interval bisection
  a0 = 0.f; a1 = 0.f;
  for (int i = tid; i < VOCAB; i += BLK) {
    uint32_t key = S[i];
    if (key >= lowKey5) { float v = k2f(key); float e = __expf(v - m); a0 += e; a1 += e * (v - m); }
  }
  const float Z5   = block_reduce_add(a0, s_wsum, tid);
  const float Sve5 = block_reduce_add(a1, s_wsum, tid);
  const float A_val = (Sve5 / Z5 - logf(Z5)) + m + logf(Z5);
  {
    const float iZ5 = 1.0f / Z5;
    float carry = 0.f;
    for (int base = 0; base < VOCAB; base += BLK) {
      int i = base + tid;
      float e = 0.f;
      if (i < VOCAB) { uint32_t key = S[i]; if (key >= lowKey5) e = __expf(k2f(key) - m) * iZ5; }
      float tot;
      float cs = carry + block_scan_inclusive(e, s_wsum, tid, &tot);
      if (i < VOCAB) Pbuf[i] = cs;
      carry += tot;
      __syncthreads();
    }
  }
  __threadfence();
  __syncthreads();

  if (wid == 0) {   // 32-lane parallel bisection on interval half-width d
    float dmax = fmaxf(fabsf(k2f(S[VOCAB - 1]) - A_val), fabsf(k2f(S[0]) - A_val));
    dmax = dmax * 1.0001f + 1e-6f;
    float dlo = 0.f, dhi = dmax;
    for (int it = 0; it < 6; ++it) {
      float dj = dlo + (dhi - dlo) * (float)(lane + 1) * (1.0f / 33.0f);
      float W  = w_interval(S, Pbuf, A_val, dj);
      unsigned long long msk = __ballot(W >= typr);
      if (msk) {
        int j0 = __ffsll((long long)msk) - 1;
        float nlo = dlo + (dhi - dlo) * (float)j0 * (1.0f / 33.0f);
        float nhi = dlo + (dhi - dlo) * (float)(j0 + 1) * (1.0f / 33.0f);
        dlo = nlo; dhi = nhi;
      } else {
        dlo = dlo + (dhi - dlo) * (32.0f / 33.0f);
      }
    }
    if (lane == 0) {
      float dkeep = dlo;
      int s5  = lb_idx(S, lowKey5);
      int lbA = lb_idx(S, f2k(A_val));
      int i1  = lbA;     if (i1 < s5) i1 = s5;      if (i1 > VOCAB - 1) i1 = VOCAB - 1;
      int i0c = lbA - 1; if (i0c < s5) i0c = s5;    if (i0c > VOCAB - 1) i0c = VOCAB - 1;
      float v1 = k2f(S[i1]), v0 = k2f(S[i0c]);
      float vc = (fabsf(v1 - A_val) < fabsf(v0 - A_val)) ? v1 : v0;   // min-dev survivor (always kept)
      uint32_t kc = f2k(vc);
      int hid = ub_idx(S, f2k(A_val + dkeep)) - 1;
      float mx2 = vc;
      if (hid >= s5) {
        float vh = k2f(S[hid]);
        if (vh >= A_val - dkeep) mx2 = fmaxf(vh, vc);
      }
      s_f[0] = dkeep; s_f[1] = mx2; s_u[0] = kc;
    }
  }
  __syncthreads();
  const float    dkeep = s_f[0];
  const float    mx2   = s_f[1];
  const uint32_t kc    = s_u[0];

  // quadratic smoothing constants
  const float ksf = ((3.0f - smcv) * 0.5f) * smf;
  const float ssf = ((smcv - 1.0f) * 0.5f) * smf;
  const float lmp = logf(minp);                 // min_p keep:  t >= log(min_p)

  // Z7 over typical & min_p survivors (transformed domain)
  acc = 0.f;
  for (int i = tid; i < VOCAB; i += BLK) {
    uint32_t key = S[i];
    if (key >= lowKey5) {
      float v = k2f(key);
      if (fabsf(v - A_val) <= dkeep || key == kc) {
        float dd = v - mx2;
        float t  = -ksf * dd * dd + ssf * dd * dd * dd;
        if (t >= lmp) acc += __expf(t);
      }
    }
  }
  const float Z7 = block_reduce_add(acc, s_wsum, tid);
  const float ta_thr = logf(topa) - logf(Z7);   // top_a keep:  t >= log(top_a) - log(Z7)

  // ---------------- Phase 5: final elementwise write -------------------------
  for (int i = tid; i < VOCAB; i += BLK) {
    float v = orow[i];
    uint32_t key = f2k(v);
    float res = NEG_INF;
    if (key >= lowKey5 && (fabsf(v - A_val) <= dkeep || key == kc)) {
      float dd = v - mx2;
      float t  = -ksf * dd * dd + ssf * dd * dd * dd;
      if (t >= lmp && t >= ta_thr) res = mx2 + t;
    }
    orow[i] = res;
  }
}

// =============================================================================
extern "C" void kernel_launch(void* const* d_in, const int* in_sizes, int n_in,
                              void* d_out, int out_size, void* d_ws, size_t ws_size,
                              hipStream_t stream) {
  (void)in_sizes; (void)n_in; (void)out_size;
  const float* logits   = (const float*)d_in[0];
  const int*   prompt   = (const int*)  d_in[1];
  const int*   outtok   = (const int*)  d_in[2];
  const float* presence = (const float*)d_in[3];
  const float* freq     = (const float*)d_in[4];
  const float* rep      = (const float*)d_in[5];
  const float* temp     = (const float*)d_in[6];
  const float* topp     = (const float*)d_in[7];
  const int*   topk     = (const int*)  d_in[8];
  const float* minp     = (const float*)d_in[9];
  const float* topa     = (const float*)d_in[10];
  const float* tfs      = (const float*)d_in[11];
  const float* typ      = (const float*)d_in[12];
  const float* epsc     = (const float*)d_in[13];
  const float* etac     = (const float*)d_in[14];
  const float* smf      = (const float*)d_in[15];
  const float* smc      = (const float*)d_in[16];
  float* out = (float*)d_out;

  const size_t need = 2ull * N_ROWS * VOCAB * sizeof(uint32_t);
  if (ws_size < need) return;   // not enough scratch for the sort ping-pong
  uint32_t* wsA = (uint32_t*)d_ws;
  uint32_t* wsB = wsA + (size_t)N_ROWS * VOCAB;

  sampler_row_kernel<<<dim3(N_ROWS), dim3(BLK), 0, stream>>>(
      logits, prompt, outtok, presence, freq, rep, temp, topp, topk,
      minp, topa, tfs, typ, epsc, etac, smf, smc, out, wsA, wsB);
}